// GATHeuristic_49950469653200
// MI455X (gfx1250) — compile-verified
//
#include <hip/hip_runtime.h>
#include <math.h>

typedef __attribute__((ext_vector_type(2))) float v2f;
typedef __attribute__((ext_vector_type(8))) float v8f;

#define H1DIM 256     // heads*hid of layer 1
#define INDIM 512
#define NSLOPE 0.2f

// ---------- float atomic max via sign-aware int trick ----------
__device__ __forceinline__ void atomicMaxFloat(float* addr, float val) {
    if (val >= 0.0f) atomicMax((int*)addr, __float_as_int(val));
    else             atomicMin((unsigned int*)addr, __float_as_uint(val));
}

// ---------- CDNA5 async global->LDS copy (ASYNCcnt path), inline asm ----------
__device__ __forceinline__ void async_copy_b128(unsigned lds_byte_off, const float* gaddr) {
    asm volatile("global_load_async_to_lds_b128 %0, %1, off"
                 :: "v"(lds_byte_off), "v"(gaddr)
                 : "memory");
}
__device__ __forceinline__ void wait_async0() {
    asm volatile("s_wait_asynccnt 0x0" ::: "memory");
}

// ---------------- init scratch ----------------
__global__ void init_kernel(float* out1, float* maxb, float* sumb,
                            float* max2, float* sum2, float* agg2, int N) {
    int i = blockIdx.x * blockDim.x + threadIdx.x;
    int stride = gridDim.x * blockDim.x;
    int total = N * H1DIM;
    for (int k = i; k < total; k += stride) out1[k] = 0.0f;
    for (int k = i; k < N * 2; k += stride) { maxb[k] = -INFINITY; sumb[k] = 0.0f; }
    for (int k = i; k < N;     k += stride) { max2[k] = -INFINITY; sum2[k] = 0.0f; agg2[k] = 0.0f; }
}

// ---------------- pre-pair W1 into Wp[kpair][n][2] so async B128 tile loads land
// ---------------- in LDS already (k,k+1)-interleaved for single ds_load_b64 B-frags
__global__ void pack_w_kernel(const float* __restrict__ W, float* __restrict__ Wp) {
    int idx = blockIdx.x * blockDim.x + threadIdx.x;   // 512*256 elements
    if (idx >= INDIM * H1DIM) return;
    int k = idx >> 8;           // /256
    int n = idx & 255;
    Wp[(size_t)(k >> 1) * (2 * H1DIM) + n * 2 + (k & 1)] = W[idx];
}

// ---------------- WMMA fp32 GEMM: Y[N x 256] = X[N x 512] * Wp ----------------
// Double-buffered LDS, async global->LDS staging overlapped with WMMA compute.
// 128 threads = 4 waves; wave tile 32x64 (acc[2][4]); block tile 128x64.
#define BM 128
#define BN 64
#define BK 32
#define KPSTRIDE 160   // dwords per LDS kpair row (128 used + 32 pad => bank-disjoint halves)

__global__ __launch_bounds__(128) void gemm_wmma(const float* __restrict__ X,
                                                 const float* __restrict__ Wp,
                                                 float* __restrict__ Y, int N) {
    __shared__ __align__(16) float As[2][BM][BK + 1];          // +1 pad -> conflict-free col reads
    __shared__ __align__(16) float BsP[2][BK / 2][KPSTRIDE];   // [kpair][n*2+par], pad 32dw

    const int tid  = threadIdx.x;
    const int lane = tid & 31;
    const int wave = tid >> 5;       // 0..3 -> M offset wave*32
    const int bm0  = blockIdx.x * BM;
    const int bn0  = blockIdx.y * BN;
    const int lm   = lane & 15;
    const int hi   = lane >> 4;      // 0/1
    const int koff = hi * 2;

    v8f acc[2][4];
    #pragma unroll
    for (int i = 0; i < 2; i++)
        #pragma unroll
        for (int j = 0; j < 4; j++) acc[i][j] = (v8f){0,0,0,0,0,0,0,0};

    // ---- issue async loads for K-chunk k0 into buffer `buf` ----
    auto issueLoads = [&](int buf, int k0) {
        // A tile: 128 rows x 32 k = 1024 B128, 8 per thread
        #pragma unroll
        for (int p = 0; p < 8; p++) {
            int idx = tid + p * 128;
            int r   = idx >> 3;            // 0..127
            int c4  = (idx & 7) * 4;
            int gr  = bm0 + r;
            if (gr < N) {   // OOB rows keep stale LDS; only feed unstored C rows
                unsigned lo = (unsigned)(size_t)&As[buf][r][c4];
                async_copy_b128(lo, X + (size_t)gr * INDIM + k0 + c4);
            }
        }
        // B tile: 16 kpairs x 64 n x 2 = 512 B128, 4 per thread
        #pragma unroll
        for (int p = 0; p < 4; p++) {
            int idx = tid + p * 128;
            int kp  = idx >> 5;            // 0..15
            int j   = idx & 31;            // b128 within kpair row (covers 2 n)
            unsigned lo = (unsigned)(size_t)&BsP[buf][kp][j * 4];
            const float* ga = Wp + (size_t)(k0 / 2 + kp) * (2 * H1DIM) + bn0 * 2 + j * 4;
            async_copy_b128(lo, ga);
        }
    };

    issueLoads(0, 0);
    wait_async0();
    __syncthreads();

    int buf = 0;
    for (int k0 = 0; k0 < INDIM; k0 += BK) {
        int nk = k0 + BK;
        if (nk < INDIM) issueLoads(buf ^ 1, nk);   // overlap next-tile DMA with compute

        #pragma unroll
        for (int kk = 0; kk < BK; kk += 4) {
            v2f a[2], b[4];
            #pragma unroll
            for (int i = 0; i < 2; i++) {
                int m = wave * 32 + i * 16 + lm;
                a[i].x = As[buf][m][kk + koff];
                a[i].y = As[buf][m][kk + koff + 1];
            }
            int kprow = (kk >> 1) + hi;            // kpair row for this lane half
            #pragma unroll
            for (int j = 0; j < 4; j++) {
                int n = j * 16 + lm;
                b[j] = *(const v2f*)&BsP[buf][kprow][n * 2];   // (k,k+1) contiguous
            }
            #pragma unroll
            for (int i = 0; i < 2; i++)
                #pragma unroll
                for (int j = 0; j < 4; j++)
                    acc[i][j] = __builtin_amdgcn_wmma_f32_16x16x4_f32(
                        false, a[i], false, b[j], (short)0, acc[i][j], false, false);
        }

        wait_async0();        // next-tile DMA done (no-op on final iter)
        __syncthreads();      // everyone finished reading `buf`
        buf ^= 1;
    }

    const int rowsel = hi * 8;
    #pragma unroll
    for (int i = 0; i < 2; i++) {
        #pragma unroll
        for (int j = 0; j < 4; j++) {
            int ncol = bn0 + j * 16 + lm;
            #pragma unroll
            for (int r = 0; r < 8; r++) {
                int grow = bm0 + wave * 32 + i * 16 + rowsel + r;
                if (grow < N) Y[(size_t)grow * H1DIM + ncol] = acc[i][j][r];
            }
        }
    }
}

// ---------------- per-node attention coefficients, layer 1 ----------------
__global__ void alpha1_kernel(const float* __restrict__ h1,
                              const float* __restrict__ att_src,
                              const float* __restrict__ att_dst,
                              float* as1, float* ad1, int N) {
    int gw   = (blockIdx.x * blockDim.x + threadIdx.x) >> 5;
    int lane = threadIdx.x & 31;
    if (gw >= N) return;
    const float* hrow = h1 + (size_t)gw * H1DIM;
    #pragma unroll
    for (int h = 0; h < 2; h++) {
        float ss = 0.f, sd = 0.f;
        #pragma unroll
        for (int i = 0; i < 4; i++) {
            int c = h * 128 + lane + i * 32;
            float v = hrow[c];
            ss += v * att_src[c];
            sd += v * att_dst[c];
        }
        #pragma unroll
        for (int off = 16; off > 0; off >>= 1) {
            ss += __shfl_down(ss, off, 32);
            sd += __shfl_down(sd, off, 32);
        }
        if (lane == 0) { as1[gw * 2 + h] = ss; ad1[gw * 2 + h] = sd; }
    }
}

// ---------------- layer-1 edge passes ----------------
__global__ void edge_max1(const int* __restrict__ ei, int E, int N,
                          const float* __restrict__ as1, const float* __restrict__ ad1,
                          float* sc1, float* maxb) {
    int t = blockIdx.x * blockDim.x + threadIdx.x;
    int Et = E + N;
    if (t >= Et * 2) return;
    int e = t >> 1, h = t & 1;
    int s = (e < E) ? ei[e]     : (e - E);
    int d = (e < E) ? ei[E + e] : (e - E);
    float sc = as1[s * 2 + h] + ad1[d * 2 + h];
    sc = sc > 0.f ? sc : NSLOPE * sc;
    sc1[(size_t)e * 2 + h] = sc;
    atomicMaxFloat(&maxb[d * 2 + h], sc);
}

__global__ void edge_exp1(const int* __restrict__ ei, int E, int N,
                          float* sc1, const float* __restrict__ maxb, float* sumb) {
    int t = blockIdx.x * blockDim.x + threadIdx.x;
    int Et = E + N;
    if (t >= Et * 2) return;
    int e = t >> 1, h = t & 1;
    int d = (e < E) ? ei[E + e] : (e - E);
    float ex = expf(sc1[(size_t)e * 2 + h] - maxb[d * 2 + h]);
    sc1[(size_t)e * 2 + h] = ex;
    atomicAdd(&sumb[d * 2 + h], ex);
}

__global__ void edge_agg1(const int* __restrict__ ei, int E, int N,
                          const float* __restrict__ h1, const float* __restrict__ sc1,
                          const float* __restrict__ sumb, float* out1) {
    int gw   = (blockIdx.x * blockDim.x + threadIdx.x) >> 5;
    int lane = threadIdx.x & 31;
    int Et = E + N;
    if (gw >= Et) return;
    int e = gw;
    int s = (e < E) ? ei[e]     : (e - E);
    int d = (e < E) ? ei[E + e] : (e - E);
    float w0 = sc1[(size_t)e * 2 + 0] / (sumb[d * 2 + 0] + 1e-16f);
    float w1 = sc1[(size_t)e * 2 + 1] / (sumb[d * 2 + 1] + 1e-16f);
    const float* hs = h1 + (size_t)s * H1DIM;
    float* od = out1 + (size_t)d * H1DIM;
    #pragma unroll
    for (int i = 0; i < 8; i++) {
        int idx = lane + i * 32;
        float w = (idx < 128) ? w0 : w1;
        atomicAdd(&od[idx], hs[idx] * w);
    }
}

// ---------------- layer-2 node kernel: h2 = elu(out1 + b1) @ W2, plus att coeffs ----------------
__global__ void node2_kernel(const float* __restrict__ out1,
                             const float* __restrict__ b1,
                             const float* __restrict__ W2,
                             const float* __restrict__ att_src2,
                             const float* __restrict__ att_dst2,
                             float* h2, float* as2, float* ad2, int N) {
    int gw   = (blockIdx.x * blockDim.x + threadIdx.x) >> 5;
    int lane = threadIdx.x & 31;
    if (gw >= N) return;
    const float* row = out1 + (size_t)gw * H1DIM;
    float s = 0.f;
    #pragma unroll
    for (int i = 0; i < 8; i++) {
        int k = lane + i * 32;
        float t = row[k] + b1[k];
        t = (t > 0.f) ? t : (expf(t) - 1.f);  // ELU(alpha=1)
        s += t * W2[k];
    }
    #pragma unroll
    for (int off = 16; off > 0; off >>= 1) s += __shfl_down(s, off, 32);
    if (lane == 0) {
        h2[gw]  = s;
        as2[gw] = s * att_src2[0];
        ad2[gw] = s * att_dst2[0];
    }
}

// ---------------- layer-2 edge passes (scalar per edge) ----------------
__global__ void edge_max2(const int* __restrict__ ei, int E, int N,
                          const float* __restrict__ as2, const float* __restrict__ ad2,
                          float* sc2, float* max2) {
    int e = blockIdx.x * blockDim.x + threadIdx.x;
    if (e >= E + N) return;
    int s = (e < E) ? ei[e]     : (e - E);
    int d = (e < E) ? ei[E + e] : (e - E);
    float sc = as2[s] + ad2[d];
    sc = sc > 0.f ? sc : NSLOPE * sc;
    sc2[e] = sc;
    atomicMaxFloat(&max2[d], sc);
}

__global__ void edge_exp2(const int* __restrict__ ei, int E, int N,
                          float* sc2, const float* __restrict__ max2, float* sum2) {
    int e = blockIdx.x * blockDim.x + threadIdx.x;
    if (e >= E + N) return;
    int d = (e < E) ? ei[E + e] : (e - E);
    float ex = expf(sc2[e] - max2[d]);
    sc2[e] = ex;
    atomicAdd(&sum2[d], ex);
}

__global__ void edge_agg2(const int* __restrict__ ei, int E, int N,
                          const float* __restrict__ h2, const float* __restrict__ sc2,
                          const float* __restrict__ sum2, float* agg2) {
    int e = blockIdx.x * blockDim.x + threadIdx.x;
    if (e >= E + N) return;
    int s = (e < E) ? ei[e]     : (e - E);
    int d = (e < E) ? ei[E + e] : (e - E);
    atomicAdd(&agg2[d], h2[s] * sc2[e] / (sum2[d] + 1e-16f));
}

__global__ void finalize_kernel(const float* __restrict__ agg2,
                                const float* __restrict__ b2,
                                float* out, int N) {
    int n = blockIdx.x * blockDim.x + threadIdx.x;
    if (n < N) out[n] = agg2[n] + b2[0];
}

extern "C" void kernel_launch(void* const* d_in, const int* in_sizes, int n_in,
                              void* d_out, int out_size, void* d_ws, size_t ws_size,
                              hipStream_t stream) {
    const float* x        = (const float*)d_in[0];
    const int*   ei       = (const int*)  d_in[1];
    const float* W1       = (const float*)d_in[2];
    const float* att_src1 = (const float*)d_in[3];
    const float* att_dst1 = (const float*)d_in[4];
    const float* b1       = (const float*)d_in[5];
    const float* W2       = (const float*)d_in[6];
    const float* att_src2 = (const float*)d_in[7];
    const float* att_dst2 = (const float*)d_in[8];
    const float* b2       = (const float*)d_in[9];
    float* out = (float*)d_out;

    const int N  = in_sizes[0] / INDIM;   // 50000
    const int E  = in_sizes[1] / 2;       // 800000
    const int Et = E + N;                 // with self loops

    // workspace carve-up (floats; every chunk is a multiple of 16 bytes)
    char* p = (char*)d_ws;
    float* h1   = (float*)p; p += (size_t)N * H1DIM * 4;
    float* out1 = (float*)p; p += (size_t)N * H1DIM * 4;
    float* Wp   = (float*)p; p += (size_t)INDIM * H1DIM * 4;
    float* as1  = (float*)p; p += (size_t)N * 2 * 4;
    float* ad1  = (float*)p; p += (size_t)N * 2 * 4;
    float* maxb = (float*)p; p += (size_t)N * 2 * 4;
    float* sumb = (float*)p; p += (size_t)N * 2 * 4;
    float* sc1  = (float*)p; p += (size_t)Et * 2 * 4;
    float* h2   = (float*)p; p += (size_t)N * 4;
    float* as2  = (float*)p; p += (size_t)N * 4;
    float* ad2  = (float*)p; p += (size_t)N * 4;
    float* max2 = (float*)p; p += (size_t)N * 4;
    float* sum2 = (float*)p; p += (size_t)N * 4;
    float* agg2 = (float*)p; p += (size_t)N * 4;
    float* sc2  = (float*)p; p += (size_t)Et * 4;

    init_kernel<<<2048, 256, 0, stream>>>(out1, maxb, sumb, max2, sum2, agg2, N);
    pack_w_kernel<<<(INDIM * H1DIM + 255) / 256, 256, 0, stream>>>(W1, Wp);

    dim3 ggrid((N + BM - 1) / BM, H1DIM / BN);
    gemm_wmma<<<ggrid, 128, 0, stream>>>(x, Wp, h1, N);

    alpha1_kernel<<<(N * 32 + 255) / 256, 256, 0, stream>>>(h1, att_src1, att_dst1, as1, ad1, N);

    edge_max1<<<(Et * 2 + 255) / 256, 256, 0, stream>>>(ei, E, N, as1, ad1, sc1, maxb);
    edge_exp1<<<(Et * 2 + 255) / 256, 256, 0, stream>>>(ei, E, N, sc1, maxb, sumb);
    edge_agg1<<<((size_t)Et * 32 + 255) / 256, 256, 0, stream>>>(ei, E, N, h1, sc1, sumb, out1);

    node2_kernel<<<(N * 32 + 255) / 256, 256, 0, stream>>>(out1, b1, W2, att_src2, att_dst2,
                                                           h2, as2, ad2, N);

    edge_max2<<<(Et + 255) / 256, 256, 0, stream>>>(ei, E, N, as2, ad2, sc2, max2);
    edge_exp2<<<(Et + 255) / 256, 256, 0, stream>>>(ei, E, N, sc2, max2, sum2);
    edge_agg2<<<(Et + 255) / 256, 256, 0, stream>>>(ei, E, N, h2, sc2, sum2, agg2);

    finalize_kernel<<<(N + 255) / 256, 256, 0, stream>>>(agg2, b2, out, N);
}